// MeanEncoder_74844100100207
// MI455X (gfx1250) — compile-verified
//
#include <hip/hip_runtime.h>
#include <stdint.h>

// ---------------------------------------------------------------------------
// Segment-mean (DGL copy_e + mean): out[n,:] = sum_{dst[e]==n} e[e,:] / cnt
// Memory-bound (~230 MB @ 23.3 TB/s ~ 10us). Strategy: coalesced b128 loads,
// no-return global_atomic_add_f32 into L2-resident output; finalize runs the
// per-row scale as D = diag(1/cnt) x S on the WMMA pipe (f32 16x16x4).
// Zero-init uses the CDNA5 async LDS->global store path (ASYNCcnt).
// ---------------------------------------------------------------------------

typedef float f4  __attribute__((ext_vector_type(4)));
typedef float v2f __attribute__((ext_vector_type(2)));
typedef float v8f __attribute__((ext_vector_type(8)));
typedef int   v4i __attribute__((vector_size(16)));   // matches async builtin param

#define AS1 __attribute__((address_space(1)))
#define AS3 __attribute__((address_space(3)))

#if defined(__has_builtin)
#  if __has_builtin(__builtin_amdgcn_global_store_async_from_lds_b128)
#    define HAVE_ASYNC_B128 1
#  endif
#  if __has_builtin(__builtin_amdgcn_s_wait_asynccnt)
#    define HAVE_WAIT_ASYNC 1
#  endif
#  if __has_builtin(__builtin_amdgcn_wmma_f32_16x16x4_f32)
#    define HAVE_WMMA_F32 1
#  endif
#endif

// --- kernel: zero a float4 region; uses CDNA5 async LDS->global store path
__global__ void me_zero_kernel(float* __restrict__ dst, long long nquads) {
  long long i = (long long)blockIdx.x * blockDim.x + threadIdx.x;
#if defined(HAVE_ASYNC_B128)
  __shared__ v4i zbuf[256];
  zbuf[threadIdx.x] = (v4i){0, 0, 0, 0};
  __syncthreads();
  if (i < nquads) {
    v4i* gp = ((v4i*)dst) + i;
    v4i* lp = &zbuf[threadIdx.x];
    // per-lane 16B DMA: LDS zeros -> global, tracked by ASYNCcnt
    __builtin_amdgcn_global_store_async_from_lds_b128(
        (AS1 v4i*)gp, (AS3 v4i*)lp, 0, 0);
  }
#  if defined(HAVE_WAIT_ASYNC)
  __builtin_amdgcn_s_wait_asynccnt(0);
#  endif
#else
  if (i < nquads) {
    ((f4*)dst)[i] = (f4)(0.0f);
  }
#endif
}

// --- kernel: detect whether dst indices are int64 (reference says int64, but
// JAX without x64 yields int32). For int64 (values < 2^31), every high dword
// is 0. False-positive probability for random int32 indices ~ (1e-5)^8.
__global__ void me_detect_kernel(const unsigned* __restrict__ dstRaw,
                                 long long E, int* __restrict__ flagOut) {
  if (blockIdx.x == 0 && threadIdx.x == 0) {
    int is64 = 1;
    int probes = (E >= 8) ? 8 : (int)E;
    for (int p = 0; p < probes; ++p) {
      if (dstRaw[2 * p + 1] != 0u) { is64 = 0; break; }
    }
    *flagOut = is64;
  }
}

// --- kernel: scatter-add. One thread per float4 of e. qshift = log2(D/4).
__global__ void me_scatter_kernel(const float* __restrict__ e,
                                  const void* __restrict__ dstRaw,
                                  const int* __restrict__ flagPtr,
                                  float* __restrict__ out,
                                  float* __restrict__ cnt,
                                  long long nquads, int qshift, int D) {
  long long qid = (long long)blockIdx.x * blockDim.x + threadIdx.x;
  if (qid >= nquads) return;
  long long edge = qid >> qshift;
  int q = (int)(qid & ((1ll << qshift) - 1));

  long long node;
  if (*flagPtr) {
    node = ((const long long*)dstRaw)[edge];
  } else {
    node = (long long)((const int*)dstRaw)[edge];
  }

  f4 v = ((const f4*)e)[qid];              // coalesced b128
  float* o = out + node * (long long)D + (q << 2);
#if defined(__HIP_DEVICE_COMPILE__)
  unsafeAtomicAdd(o + 0, v.x);             // native global_atomic_add_f32,
  unsafeAtomicAdd(o + 1, v.y);             // no-return (STOREcnt path)
  unsafeAtomicAdd(o + 2, v.z);
  unsafeAtomicAdd(o + 3, v.w);
  if (q == 0) unsafeAtomicAdd(cnt + node, 1.0f);
#else
  atomicAdd(o + 0, v.x);
  atomicAdd(o + 1, v.y);
  atomicAdd(o + 2, v.z);
  atomicAdd(o + 3, v.w);
  if (q == 0) atomicAdd(cnt + node, 1.0f);
#endif
}

// --- kernel: WMMA finalize. One wave per 16-node x 16-feature tile:
//   out_tile = diag(1/max(cnt,1)) x sums_tile  via 4 chained
//   V_WMMA_F32_16X16X4_F32 (K=16 contraction; off-diagonal terms exactly 0).
// Layouts (wave32):
//   A 16x4 f32 : lane l -> M=l&15;           reg r -> K = 2*(l>>4) + r
//   B 4x16 f32 : lane l -> N=l&15;           reg r -> K = 2*(l>>4) + r
//   C/D 16x16  : lane l -> N=l&15;           reg r -> M = r + 8*(l>>4)
// Falls back to an equivalent scalar tile body if the builtin is absent.
__global__ void me_finalize_wmma_kernel(float* __restrict__ out,
                                        const float* __restrict__ cnt,
                                        int nTilesM, int nTilesF, int D) {
  int wave = (int)(((long long)blockIdx.x * blockDim.x + threadIdx.x) >> 5);
  int lane = (int)(threadIdx.x & 31);
  int total = nTilesM * nTilesF;
  if (wave >= total) return;               // wave-uniform; EXEC stays all-1s
  int tm = wave / nTilesF;
  int tf = wave - tm * nTilesF;
  long long nb = (long long)tm * 16;       // node base (tile rows)
  int fb = tf * 16;                        // feature base (tile cols)
  int h  = lane >> 4;                      // lane half
  int ln = lane & 15;

#if defined(HAVE_WMMA_F32)
  float c0  = cnt[nb + ln];
  float inv = 1.0f / fmaxf(c0, 1.0f);

  v8f acc = {};
#pragma unroll
  for (int kb = 0; kb < 4; ++kb) {
    int k0 = kb * 4 + 2 * h;               // K of reg0 for this lane-half
    v2f a, b;
    a.x = (ln == k0)     ? inv : 0.0f;     // diag(inv) slice, cols k0..k0+1
    a.y = (ln == k0 + 1) ? inv : 0.0f;
    const float* brow = out + (nb + k0) * (long long)D + fb + ln;
    b.x = brow[0];                          // S row k0
    b.y = brow[D];                          // S row k0+1
    acc = __builtin_amdgcn_wmma_f32_16x16x4_f32(
        /*neg_a=*/false, a, /*neg_b=*/false, b,
        /*c_mod=*/(short)0, acc, /*reuse_a=*/false, /*reuse_b=*/false);
  }
#pragma unroll
  for (int r = 0; r < 8; ++r) {
    out[(nb + r + 8 * h) * (long long)D + fb + ln] = acc[r];
  }
#else
  // scalar tile body with identical work assignment
#pragma unroll
  for (int r = 0; r < 8; ++r) {
    long long m = nb + r + 8 * h;
    float denom = fmaxf(cnt[m], 1.0f);
    long long idx = m * (long long)D + fb + ln;
    out[idx] = out[idx] / denom;
  }
#endif
}

// --- kernel: scalar finalize fallback (general shapes).
__global__ void me_finalize_kernel(float* __restrict__ out,
                                   const float* __restrict__ cnt,
                                   long long nquads, int fshift) {
  long long i = (long long)blockIdx.x * blockDim.x + threadIdx.x;
  if (i >= nquads) return;
  long long n = i >> fshift;
  float denom = fmaxf(cnt[n], 1.0f);
  f4 v = ((f4*)out)[i];
  v.x = v.x / denom;
  v.y = v.y / denom;
  v.z = v.z / denom;
  v.w = v.w / denom;
  ((f4*)out)[i] = v;
}

static inline unsigned me_blocks(long long work, int threads) {
  return (unsigned)((work + threads - 1) / threads);
}

static inline int me_log2i(long long v) {
  int s = 0;
  while ((1ll << s) < v) ++s;
  return s;
}

extern "C" void kernel_launch(void* const* d_in, const int* in_sizes, int n_in,
                              void* d_out, int out_size, void* d_ws, size_t ws_size,
                              hipStream_t stream) {
  const float* e       = (const float*)d_in[0];
  const void*  dstRaw  = d_in[1];
  long long E = (long long)in_sizes[1];
  long long eElems = (long long)in_sizes[0];
  int D = (int)(eElems / E);                 // 32 in the reference
  int N = out_size / D;                      // 100000

  float* out = (float*)d_out;
  float* cnt = (float*)d_ws;
  size_t cntBytes = ((size_t)N * sizeof(float) + 255) & ~(size_t)255;
  int* flag = (int*)((char*)d_ws + cntBytes);

  const int T = 256;
  long long outQ = (long long)out_size / 4;          // D%4==0 (D=32)
  long long cntQ = ((long long)N + 3) / 4;

  // 1) zero accumulators (async LDS->global store path)
  me_zero_kernel<<<me_blocks(outQ, T), T, 0, stream>>>(out, outQ);
  me_zero_kernel<<<me_blocks(cntQ, T), T, 0, stream>>>(cnt, cntQ);

  // 2) sniff dst dtype (int64 vs int32)
  me_detect_kernel<<<1, 32, 0, stream>>>((const unsigned*)dstRaw, E, flag);

  // 3) scatter-add sums and counts
  int quadsPerEdge = D / 4;                  // 8
  int qshift = me_log2i(quadsPerEdge);       // 3
  long long scatQ = E * (long long)quadsPerEdge;
  me_scatter_kernel<<<me_blocks(scatQ, T), T, 0, stream>>>(
      e, dstRaw, flag, out, cnt, scatQ, qshift, D);

  // 4) divide by max(count,1): WMMA diag-scale when tiles fit, else scalar
  if ((N % 16 == 0) && (D % 16 == 0)) {
    int nTilesM = N / 16;
    int nTilesF = D / 16;
    long long waves = (long long)nTilesM * nTilesF;
    me_finalize_wmma_kernel<<<me_blocks(waves * 32, T), T, 0, stream>>>(
        out, cnt, nTilesM, nTilesF, D);
  } else {
    me_finalize_kernel<<<me_blocks(outQ, T), T, 0, stream>>>(out, cnt, outQ,
                                                             qshift);
  }
}